// HeatmapGenerator_40415642255883
// MI455X (gfx1250) — compile-verified
//
#include <hip/hip_runtime.h>

typedef float v2f __attribute__((ext_vector_type(2)));
typedef float v8f __attribute__((ext_vector_type(8)));

#define HM        112
#define TILES     7                       // 112 / 16
#define INV_2S2   (1.0f / 8.0f)           // 1 / (2 * sigma^2), sigma = 2

// One block per landmark: 224 threads = 7 wave32s, wave w owns tile-row w.
// Phase 1: separable gaussian factor tables ex[112], ey[112] in LDS
//          (224 exps instead of 12544); peak is analytic (nearest in-range
//          integer per axis), and 1/(peak+1e-6) is folded into ey.
// Phase 2: each 16x16 output tile is a rank-1 outer product computed by one
//          V_WMMA_F32_16X16X4_F32 (K=0 column of A = ey slice, K=0 row of
//          B = ex slice), streamed out with non-temporal b32 stores
//          (1.04 GB output >> 192 MB L2).
__global__ __launch_bounds__(224) void heatmap_wmma_kernel(
    const float* __restrict__ lm,   // [npts, 2] normalized (x, y)
    float* __restrict__ out,        // [npts, 112, 112]
    int npts)
{
    __shared__ float sex[HM];
    __shared__ float sey[HM];

    const int p = blockIdx.x;
    if (p >= npts) return;

    const int tid  = threadIdx.x;
    const int lane = tid & 31;
    const int wave = tid >> 5;      // 0..6 == tile row

    // Landmark in pixel coords.
    const float lx = lm[2 * p + 0] * (float)HM;
    const float ly = lm[2 * p + 1] * (float)HM;

    // Peak over the integer grid: product of per-axis maxima, each attained
    // at the nearest in-range integer (ties give equal values).
    const float dxm = fminf(fmaxf(floorf(lx + 0.5f), 0.0f), (float)(HM - 1)) - lx;
    const float dym = fminf(fmaxf(floorf(ly + 0.5f), 0.0f), (float)(HM - 1)) - ly;
    const float peak = __expf(-(dxm * dxm) * INV_2S2) * __expf(-(dym * dym) * INV_2S2);
    const float inv_peak = 1.0f / (peak + 1e-6f);

    // Phase 1: threads 0..111 -> ex, threads 112..223 -> ey (all threads used).
    if (tid < HM) {
        const float d = (float)tid - lx;
        sex[tid] = __expf(-(d * d) * INV_2S2);
    } else {
        const int w = tid - HM;     // 0..111
        const float d = (float)w - ly;
        sey[w] = __expf(-(d * d) * INV_2S2) * inv_peak;  // fold normalization in
    }
    __syncthreads();

    const int half = lane >> 4;     // 0: lanes 0-15, 1: lanes 16-31
    const int l15  = lane & 15;
    const int ti   = wave;          // this wave's tile row

    // A (16x4, MxK): lanes 0-15 of VGPR0 hold K=0 -> ey slice (loop invariant).
    // Unconditional LDS read (lanes 16-31 broadcast-read), then select.
    const float ayv = sey[ti * 16 + l15];
    v2f a;
    a.x = half ? 0.0f : ayv;
    a.y = 0.0f;

    // Output base for this wave's 16 rows: lane covers column l15 of rows
    // row0..row0+7 (D layout: VGPR r, lane L -> M = r + 8*(L>=16), N = L%16).
    const int row0 = ti * 16 + half * 8;
    float* po = out + (size_t)p * (size_t)(HM * HM) + (size_t)row0 * HM + l15;

#pragma unroll
    for (int tj = 0; tj < TILES; ++tj) {
        // B (4x16, KxN): lanes 0-15 of VGPR0 hold row K=0 -> ex slice.
        const float bxv = sex[tj * 16 + l15];
        v2f b;
        b.x = half ? 0.0f : bxv;
        b.y = 0.0f;

        v8f c = {};
        // (neg_a, A, neg_b, B, c_mod, C, reuse_a, reuse_b)
        c = __builtin_amdgcn_wmma_f32_16x16x4_f32(
            false, a, false, b, (short)0, c, false, false);

#pragma unroll
        for (int r = 0; r < 8; ++r) {
            __builtin_nontemporal_store(c[r], po + r * HM + tj * 16);
        }
    }
}

extern "C" void kernel_launch(void* const* d_in, const int* in_sizes, int n_in,
                              void* d_out, int out_size, void* d_ws, size_t ws_size,
                              hipStream_t stream) {
    (void)n_in; (void)out_size; (void)d_ws; (void)ws_size;
    const float* lm = (const float*)d_in[0];
    float* out = (float*)d_out;
    const int npts = in_sizes[0] / 2;   // 256 * 81 = 20736 landmarks
    heatmap_wmma_kernel<<<npts, 224, 0, stream>>>(lm, out, npts);
}